// DNN_WPE_24464133718529
// MI455X (gfx1250) — compile-verified
//
#include <hip/hip_runtime.h>

typedef float v2f __attribute__((ext_vector_type(2)));
typedef float v8f __attribute__((ext_vector_type(8)));

#define B_    8
#define T_    600
#define C_    8
#define F_    257
#define TAPS  5
#define DELAY 3
#define K_    40          // TAPS*C_
#define PAD   7           // DELAY+TAPS-1
#define TP    608         // padded time length (>= T_+PAD, rounded)
#define TPR   612         // LDS row stride (612 % 64 = 36 -> conflict-free)
#define NBF   (B_*F_)     // 2056

// ---------------------------------------------------------------------------
// Kernel 1: (B,T,C,F) -> (B,F,C,T) transpose of re/im
// ---------------------------------------------------------------------------
__global__ __launch_bounds__(256) void wpe_transpose(
    const float* __restrict__ dre, const float* __restrict__ dim,
    float* __restrict__ yre, float* __restrict__ yim) {
  long idx = (long)blockIdx.x * 256 + threadIdx.x;
  const long total = (long)B_ * T_ * C_ * F_;
  if (idx >= total) return;
  int f = (int)(idx % F_); long r = idx / F_;
  int c = (int)(r % C_);   r /= C_;
  int t = (int)(r % T_);   int b = (int)(r / T_);
  long dst = ((long)(b * F_ + f) * C_ + c) * T_ + t;
  yre[dst] = dre[idx];
  yim[dst] = dim[idx];
}

// ---------------------------------------------------------------------------
// Kernel 2: inverse power  invp[(b*F+f)*T + t] = 1/max(mean_c |Y|^2, 1e-10)
// ---------------------------------------------------------------------------
__global__ __launch_bounds__(256) void wpe_invpow(
    const float* __restrict__ dre, const float* __restrict__ dim,
    float* __restrict__ invp) {
  int bt = blockIdx.x; int b = bt / T_; int t = bt % T_;
  for (int f = threadIdx.x; f < F_; f += blockDim.x) {
    long base = (long)(b * T_ + t) * C_ * F_ + f;
    float s = 0.f;
#pragma unroll
    for (int c = 0; c < C_; ++c) {
      float re = dre[base + (long)c * F_];
      float im = dim[base + (long)c * F_];
      s += re * re + im * im;
    }
    s *= (1.0f / C_);
    s = fmaxf(s, 1e-10f);
    invp[(long)(b * F_ + f) * T_ + t] = 1.0f / s;
  }
}

// ---------------------------------------------------------------------------
// WMMA helper: D = A(16x4) * B(4x16) + C, fp32
// ---------------------------------------------------------------------------
__device__ __forceinline__ v8f wmma4(v2f a, v2f b, v8f c) {
  return __builtin_amdgcn_wmma_f32_16x16x4_f32(
      false, a, false, b, (short)0, c, false, false);
}

// ---------------------------------------------------------------------------
// Kernel 3: per-(b,f) correlation GEMMs with fp32 WMMA.
//   R[k,l] = sum_t Ytw[k,t] * conj(Yt[l,t])   (40x40 complex, padded to 48)
//   P[k,c] = sum_t Ytw[k,t] * conj(Y[c,t])    (40x8 complex)
// Yt[k,t] = Y[k%8, t - DELAY - k/8] (zero padded); Ytw = Yt * invp[t].
// LDS rows 0..7 hold the padded signal (L[c][j] = Y[c][j-PAD]); row 8 is a
// zero guard row so out-of-range lanes (k>=40, n>=8) load zeros branch-free
// and EXEC stays all-ones around every WMMA.
// 4 waves: waves 0..2 -> R row-tiles 0..2 (3 col tiles each); wave 3 -> P.
// ---------------------------------------------------------------------------
__global__ __launch_bounds__(128) void wpe_corr(
    const float* __restrict__ yre, const float* __restrict__ yim,
    const float* __restrict__ invp,
    float* __restrict__ Rws, float* __restrict__ Pws) {
  __shared__ float lre[C_ + 1][TPR];
  __shared__ float lim[C_ + 1][TPR];
  __shared__ float lip[TP];

  const int bf = blockIdx.x;
  const float* gyr = yre + (long)bf * C_ * T_;
  const float* gyi = yim + (long)bf * C_ * T_;
  const float* gip = invp + (long)bf * T_;

  for (int c = 0; c < C_; ++c) {
    for (int j = threadIdx.x; j < TPR; j += blockDim.x) {
      int t = j - PAD;
      float vr = 0.f, vi = 0.f;
      if (t >= 0 && t < T_) { vr = gyr[c * T_ + t]; vi = gyi[c * T_ + t]; }
      lre[c][j] = vr; lim[c][j] = vi;
    }
  }
  for (int j = threadIdx.x; j < TPR; j += blockDim.x) {  // zero guard row
    lre[C_][j] = 0.f; lim[C_][j] = 0.f;
  }
  for (int j = threadIdx.x; j < TP; j += blockDim.x)
    lip[j] = (j < T_) ? gip[j] : 0.f;
  __syncthreads();

  const int lane = threadIdx.x & 31;
  const int wave = threadIdx.x >> 5;
  const int m = lane & 15;      // row of A / col of B (per-lane)
  const int h = lane >> 4;      // lane-half selects K 0/1 vs 2/3

  const v8f vz = {0.f, 0.f, 0.f, 0.f, 0.f, 0.f, 0.f, 0.f};

  if (wave < 3) {
    // ---- R row-tile r, three 16x16 column tiles ----
    const int r = wave;
    const int kA = 16 * r + m;
    const bool va = (kA < K_);
    const int cA = va ? (kA & 7) : C_;      // invalid -> zero guard row
    const int tapA = va ? (kA >> 3) : 0;

    int cB[3], tapB[3];
#pragma unroll
    for (int ct = 0; ct < 3; ++ct) {
      int kB = 16 * ct + m;
      bool vb = (kB < K_);
      cB[ct] = vb ? (kB & 7) : C_;
      tapB[ct] = vb ? (kB >> 3) : 0;
    }

    v8f accRe[3], accIm[3];
#pragma unroll
    for (int ct = 0; ct < 3; ++ct) { accRe[ct] = vz; accIm[ct] = vz; }

#pragma unroll 2
    for (int t0 = 0; t0 < T_; t0 += 4) {
      const int ta = t0 + 2 * h;            // this lane-half covers K = 2h, 2h+1
      const float ip0 = lip[ta], ip1 = lip[ta + 1];
      const int jA = ta + 4 - tapA;
      v2f aRe, aIm;
      aRe.x = lre[cA][jA] * ip0;  aRe.y = lre[cA][jA + 1] * ip1;
      aIm.x = lim[cA][jA] * ip0;  aIm.y = lim[cA][jA + 1] * ip1;
#pragma unroll
      for (int ct = 0; ct < 3; ++ct) {
        const int jB = ta + 4 - tapB[ct];
        v2f bRe, bIm;
        bRe.x = lre[cB[ct]][jB];  bRe.y = lre[cB[ct]][jB + 1];
        bIm.x = lim[cB[ct]][jB];  bIm.y = lim[cB[ct]][jB + 1];
        v2f bNim = -bIm;                    // conj(B): f32 WMMA has no B-neg
        accRe[ct] = wmma4(aRe, bRe,  accRe[ct]);
        accRe[ct] = wmma4(aIm, bIm,  accRe[ct]);
        accIm[ct] = wmma4(aIm, bRe,  accIm[ct]);
        accIm[ct] = wmma4(aRe, bNim, accIm[ct]);
      }
    }
    // store: acc vgpr v holds row M = 8*h + v, col N = m (ISA 7.12.2)
#pragma unroll
    for (int ct = 0; ct < 3; ++ct) {
#pragma unroll
      for (int v = 0; v < 8; ++v) {
        int row = 16 * r + 8 * h + v;
        int col = 16 * ct + m;
        if (row < K_ && col < K_) {
          long o = ((long)bf * K_ * K_ + (long)row * K_ + col) * 2;
          Rws[o] = accRe[ct][v]; Rws[o + 1] = accIm[ct][v];
        }
      }
    }
  } else {
    // ---- P: three row tiles vs one 16-wide column tile (channels 0..7) ----
    int cA[3], tapA[3];
#pragma unroll
    for (int r = 0; r < 3; ++r) {
      int kA = 16 * r + m;
      bool va = (kA < K_);
      cA[r] = va ? (kA & 7) : C_;
      tapA[r] = va ? (kA >> 3) : 0;
    }
    const int cN = (m < C_) ? m : C_;       // invalid cols -> zero guard row

    v8f pRe[3], pIm[3];
#pragma unroll
    for (int r = 0; r < 3; ++r) { pRe[r] = vz; pIm[r] = vz; }

#pragma unroll 2
    for (int t0 = 0; t0 < T_; t0 += 4) {
      const int ta = t0 + 2 * h;
      const float ip0 = lip[ta], ip1 = lip[ta + 1];
      v2f bRe, bIm;
      bRe.x = lre[cN][ta + PAD];  bRe.y = lre[cN][ta + PAD + 1];
      bIm.x = lim[cN][ta + PAD];  bIm.y = lim[cN][ta + PAD + 1];
      v2f bNim = -bIm;
#pragma unroll
      for (int r = 0; r < 3; ++r) {
        const int jA = ta + 4 - tapA[r];
        v2f aRe, aIm;
        aRe.x = lre[cA[r]][jA] * ip0;  aRe.y = lre[cA[r]][jA + 1] * ip1;
        aIm.x = lim[cA[r]][jA] * ip0;  aIm.y = lim[cA[r]][jA + 1] * ip1;
        pRe[r] = wmma4(aRe, bRe,  pRe[r]);
        pRe[r] = wmma4(aIm, bIm,  pRe[r]);
        pIm[r] = wmma4(aIm, bRe,  pIm[r]);
        pIm[r] = wmma4(aRe, bNim, pIm[r]);
      }
    }
#pragma unroll
    for (int r = 0; r < 3; ++r) {
#pragma unroll
      for (int v = 0; v < 8; ++v) {
        int row = 16 * r + 8 * h + v;
        if (row < K_ && m < C_) {
          long o = ((long)bf * K_ * C_ + (long)row * C_ + m) * 2;
          Pws[o] = pRe[r][v]; Pws[o + 1] = pIm[r][v];
        }
      }
    }
  }
}

// ---------------------------------------------------------------------------
// Kernel 4: regularize + complex Gaussian elimination (Hermitian PD, no
// pivoting) + back-substitution. One block per (b,f).
// ---------------------------------------------------------------------------
__global__ __launch_bounds__(64) void wpe_solve(
    const float* __restrict__ Rws, const float* __restrict__ Pws,
    float* __restrict__ Gws) {
  __shared__ float sRr[K_ * K_], sRi[K_ * K_];
  __shared__ float sPr[K_ * C_], sPi[K_ * C_];
  __shared__ float s_eps;

  const int bf = blockIdx.x;
  for (int i = threadIdx.x; i < K_ * K_; i += blockDim.x) {
    long o = ((long)bf * K_ * K_ + i) * 2;
    sRr[i] = Rws[o]; sRi[i] = Rws[o + 1];
  }
  for (int i = threadIdx.x; i < K_ * C_; i += blockDim.x) {
    long o = ((long)bf * K_ * C_ + i) * 2;
    sPr[i] = Pws[o]; sPi[i] = Pws[o + 1];
  }
  __syncthreads();

  if (threadIdx.x == 0) {
    float tr = 0.f;
    for (int i = 0; i < K_; ++i) tr += sRr[i * K_ + i];
    s_eps = 1e-7f * tr / K_ + 1e-10f;
  }
  __syncthreads();
  if (threadIdx.x < K_) sRr[threadIdx.x * K_ + threadIdx.x] += s_eps;
  __syncthreads();

  // forward elimination
  for (int i = 0; i < K_; ++i) {
    float pr = sRr[i * K_ + i], pi = sRi[i * K_ + i];
    float inv = 1.0f / (pr * pr + pi * pi);
    float ir = pr * inv, ii = -pi * inv;     // 1/pivot
    int j = i + 1 + threadIdx.x;
    if (j < K_) {
      float xr = sRr[j * K_ + i], xi = sRi[j * K_ + i];
      float fr = xr * ir - xi * ii;
      float fi = xr * ii + xi * ir;
      for (int k = i; k < K_; ++k) {
        float ar = sRr[i * K_ + k], ai = sRi[i * K_ + k];
        sRr[j * K_ + k] -= fr * ar - fi * ai;
        sRi[j * K_ + k] -= fr * ai + fi * ar;
      }
      for (int c = 0; c < C_; ++c) {
        float ar = sPr[i * C_ + c], ai = sPi[i * C_ + c];
        sPr[j * C_ + c] -= fr * ar - fi * ai;
        sPi[j * C_ + c] -= fr * ai + fi * ar;
      }
    }
    __syncthreads();
  }

  // back substitution (G overwrites P)
  for (int i = K_ - 1; i >= 0; --i) {
    int c = threadIdx.x;
    if (c < C_) {
      float sr = sPr[i * C_ + c], si = sPi[i * C_ + c];
      for (int j = i + 1; j < K_; ++j) {
        float ar = sRr[i * K_ + j], ai = sRi[i * K_ + j];
        float gr = sPr[j * C_ + c], gi = sPi[j * C_ + c];
        sr -= ar * gr - ai * gi;
        si -= ar * gi + ai * gr;
      }
      float pr = sRr[i * K_ + i], pi = sRi[i * K_ + i];
      float inv = 1.0f / (pr * pr + pi * pi);
      sPr[i * C_ + c] = (sr * pr + si * pi) * inv;
      sPi[i * C_ + c] = (si * pr - sr * pi) * inv;
    }
    __syncthreads();
  }

  for (int i = threadIdx.x; i < K_ * C_; i += blockDim.x) {
    long o = ((long)bf * K_ * C_ + i) * 2;
    Gws[o] = sPr[i]; Gws[o + 1] = sPi[i];
  }
}

// ---------------------------------------------------------------------------
// Kernel 5: enhanced = Y - G^H Yt, length mask, write (B,T,C,F) re||im.
// ---------------------------------------------------------------------------
__global__ __launch_bounds__(256) void wpe_apply(
    const float* __restrict__ yre, const float* __restrict__ yim,
    const float* __restrict__ Gws, const int* __restrict__ ilens,
    float* __restrict__ out) {
  __shared__ float lre[C_][TP];
  __shared__ float lim[C_][TP];
  __shared__ float gr[K_ * C_], gi[K_ * C_];

  const int bf = blockIdx.x;
  const int b = bf / F_, f = bf % F_;
  const float* gyr = yre + (long)bf * C_ * T_;
  const float* gyi = yim + (long)bf * C_ * T_;

  for (int c = 0; c < C_; ++c) {
    for (int j = threadIdx.x; j < TP; j += blockDim.x) {
      int t = j - PAD;
      float vr = 0.f, vi = 0.f;
      if (t >= 0 && t < T_) { vr = gyr[c * T_ + t]; vi = gyi[c * T_ + t]; }
      lre[c][j] = vr; lim[c][j] = vi;
    }
  }
  for (int i = threadIdx.x; i < K_ * C_; i += blockDim.x) {
    long o = ((long)bf * K_ * C_ + i) * 2;
    gr[i] = Gws[o]; gi[i] = Gws[o + 1];
  }
  __syncthreads();

  const int len = ilens[b];
  const long imOff = (long)B_ * T_ * C_ * F_;

  for (int e = threadIdx.x; e < C_ * T_; e += blockDim.x) {
    int t = e % T_, c = e / T_;
    float er = 0.f, ei = 0.f;
    if (t < len) {
      er = lre[c][t + PAD]; ei = lim[c][t + PAD];
#pragma unroll
      for (int tap = 0; tap < TAPS; ++tap) {
        int j = t + 4 - tap;
#pragma unroll
        for (int ck = 0; ck < C_; ++ck) {
          int k = tap * C_ + ck;
          float grv = gr[k * C_ + c], giv = gi[k * C_ + c];
          float yr = lre[ck][j], yi = lim[ck][j];
          er -= grv * yr + giv * yi;   // conj(G)*Yt, real
          ei -= grv * yi - giv * yr;   // conj(G)*Yt, imag
        }
      }
    }
    long o = ((long)(b * T_ + t) * C_ + c) * F_ + f;
    out[o] = er;
    out[imOff + o] = ei;
  }
}

// ---------------------------------------------------------------------------
extern "C" void kernel_launch(void* const* d_in, const int* in_sizes, int n_in,
                              void* d_out, int out_size, void* d_ws, size_t ws_size,
                              hipStream_t stream) {
  const float* dre = (const float*)d_in[0];
  const float* dim = (const float*)d_in[1];
  const int* ilens = (const int*)d_in[2];
  float* out = (float*)d_out;

  const long NY = (long)B_ * F_ * C_ * T_;       // 9,868,800
  const long NI = (long)B_ * F_ * T_;            // 1,233,600
  const long NR = (long)NBF * K_ * K_ * 2;       // 6,579,200
  const long NP = (long)NBF * K_ * C_ * 2;       // 1,315,840

  float* ws  = (float*)d_ws;
  float* yre = ws;
  float* yim = yre + NY;
  float* invp = yim + NY;
  float* Rws = invp + NI;
  float* Pws = Rws + NR;
  float* Gws = Pws + NP;

  {
    long total = (long)B_ * T_ * C_ * F_;
    int blocks = (int)((total + 255) / 256);
    wpe_transpose<<<blocks, 256, 0, stream>>>(dre, dim, yre, yim);
  }
  wpe_invpow<<<B_ * T_, 256, 0, stream>>>(dre, dim, invp);
  wpe_corr<<<NBF, 128, 0, stream>>>(yre, yim, invp, Rws, Pws);
  wpe_solve<<<NBF, 64, 0, stream>>>(Rws, Pws, Gws);
  wpe_apply<<<NBF, 256, 0, stream>>>(yre, yim, Gws, ilens, out);
}